// SemanticProcessor_11029476016562
// MI455X (gfx1250) — compile-verified
//
#include <hip/hip_runtime.h>
#include <math.h>

#define TT   192
#define BB   2
#define HH   256
#define NHH  8
#define HDD  32
#define NEE  8
#define ROWS (BB*TT)        /* 384   */
#define NPOS (BB*TT*TT)     /* 73728 */

static constexpr float INV_SCALE = 0.17677669529663687f; /* 1/sqrt(32) */

typedef __attribute__((ext_vector_type(2))) float v2f;
typedef __attribute__((ext_vector_type(8))) float v8f;

__device__ __forceinline__ float wred_sum(float v){
  v += __shfl_xor(v, 16, 32);
  v += __shfl_xor(v,  8, 32);
  v += __shfl_xor(v,  4, 32);
  v += __shfl_xor(v,  2, 32);
  v += __shfl_xor(v,  1, 32);
  return v;
}
__device__ __forceinline__ float wred_max(float v){
  v = fmaxf(v, __shfl_xor(v, 16, 32));
  v = fmaxf(v, __shfl_xor(v,  8, 32));
  v = fmaxf(v, __shfl_xor(v,  4, 32));
  v = fmaxf(v, __shfl_xor(v,  2, 32));
  v = fmaxf(v, __shfl_xor(v,  1, 32));
  return v;
}
__device__ __forceinline__ float gelu_exact(float x){
  return 0.5f * x * (1.0f + erff(x * 0.7071067811865475f));
}

/* ---------------- layernorm: one wave per row ---------------- */
__global__ void k_layernorm(const float* __restrict__ in, const float* __restrict__ g,
                            const float* __restrict__ b, float* __restrict__ out,
                            int rows, int cols){
  int row = blockIdx.x * blockDim.y + threadIdx.y;
  if (row >= rows) return;
  int lane = threadIdx.x;
  const float* r = in + (size_t)row * cols;
  float s = 0.f, ss = 0.f;
  for (int c = lane; c < cols; c += 32){ float v = r[c]; s += v; ss += v*v; }
  s = wred_sum(s); ss = wred_sum(ss);
  float mu = s / (float)cols;
  float rstd = rsqrtf(ss / (float)cols - mu*mu + 1e-5f);
  float* o = out + (size_t)row * cols;
  for (int c = lane; c < cols; c += 32) o[c] = (r[c] - mu) * rstd * g[c] + b[c];
}

/* ---------------- generic f32 WMMA GEMM, 16x16 tile per wave ----------------
   C[M,N] = A[M,K] @ W[K,N] + bias ; act==1 -> relu
   V_WMMA_F32_16X16X4_F32 layouts:
     A(16x4): lane<16 rows m=lane, K = k0+{0,1}; lanes 16-31 same rows, K = k0+{2,3}
     B(4x16): lane&15 = n, half picks K pair (mirrors A)
     C/D:     vgpr i -> M = i + 8*half, N = lane&15                           */
__global__ void __launch_bounds__(32) k_gemm16(
    const float* __restrict__ A, const float* __restrict__ W,
    const float* __restrict__ bias, float* __restrict__ C,
    int M, int N, int K, int act){
  int lane = threadIdx.x;
  int half = lane >> 4, mr = lane & 15;
  int tn = blockIdx.x * 16, tm = blockIdx.y * 16;
  v8f acc = {};
  const float* arow = A + (size_t)(tm + mr) * K;
  for (int k0 = 0; k0 < K; k0 += 4){
    int kk = k0 + 2*half;
    v2f a = *(const v2f*)(arow + kk);
    v2f b;
    b.x = W[(size_t)kk       * N + tn + mr];
    b.y = W[(size_t)(kk + 1) * N + tn + mr];
    acc = __builtin_amdgcn_wmma_f32_16x16x4_f32(false, a, false, b, (short)0, acc, false, false);
  }
  for (int i = 0; i < 8; i++){
    int row = tm + i + 8*half;
    int col = tn + mr;
    float v = acc[i] + (bias ? bias[col] : 0.f);
    if (act == 1) v = fmaxf(v, 0.f);
    C[(size_t)row * N + col] = v;
  }
}

/* ---------------- fused attention (opt. edge bias), one wave per (b,h,t) ----- */
__global__ void __launch_bounds__(32) k_attn(
    const float* __restrict__ qkv, float* __restrict__ out,
    const float* __restrict__ curb, const float* __restrict__ edges){
  __shared__ float sc[TT];
  int t = blockIdx.x, h = blockIdx.y, b = blockIdx.z;
  int d = threadIdx.x;
  int ch = h * HDD + d;
  const float* qrow = qkv + (size_t)(b*TT + t) * (3*HH);
  float qd = qrow[ch];
  float cd = 0.f;
  const float* erow = nullptr;
  if (edges){
    cd = curb[(size_t)(b*TT + t) * HH + ch] * INV_SCALE;
    erow = edges + (size_t)(b*TT + t) * TT * HH;
  }
  for (int s = 0; s < TT; s++){
    float kd = qkv[(size_t)(b*TT + s) * (3*HH) + HH + ch];
    float p = qd * kd * INV_SCALE;
    if (edges) p += cd * erow[(size_t)s * HH + ch];
    p = wred_sum(p);
    if (d == 0) sc[s] = p;
  }
  __syncthreads();
  float mx = -1e30f;
  for (int s = d; s < TT; s += 32) mx = fmaxf(mx, sc[s]);
  mx = wred_max(mx);
  float se = 0.f;
  for (int s = d; s < TT; s += 32) se += expf(sc[s] - mx);
  se = wred_sum(se);
  float inv = 1.f / se;
  float o = 0.f;
  for (int s = 0; s < TT; s++){
    float a = expf(sc[s] - mx) * inv;
    o += a * qkv[(size_t)(b*TT + s) * (3*HH) + 2*HH + ch];
  }
  out[(size_t)(b*TT + t) * HH + ch] = o;
}

/* ---------------- edge-scorer ln0 stats per position ---------------- */
__global__ void k_edge_stats(const float* __restrict__ concepts, float* __restrict__ stats){
  int pos = blockIdx.x * blockDim.y + threadIdx.y;
  if (pos >= NPOS) return;
  int lane = threadIdx.x;
  int b = pos / (TT*TT); int r = pos % (TT*TT); int t = r / TT; int s = r % TT;
  const float* q  = concepts + (size_t)(b*TT + t) * HH;
  const float* kx = concepts + (size_t)(b*TT + s) * HH;
  float sm = 0.f, ss = 0.f;
  for (int c = lane; c < HH; c += 32){
    float qv = q[c], kv = kx[c], pv = qv * kv * INV_SCALE;
    sm += qv + kv + pv;
    ss += qv*qv + kv*kv + pv*pv;
  }
  sm = wred_sum(sm); ss = wred_sum(ss);
  float mu = sm / 768.f;
  stats[2*pos]   = mu;
  stats[2*pos+1] = rsqrtf(ss / 768.f - mu*mu + 1e-5f);
}

/* fold ln0 gamma into W1; column sums for the mean/beta terms */
__global__ void k_prep_w1(const float* __restrict__ W1, const float* __restrict__ g0,
                          const float* __restrict__ b0, const float* __restrict__ b1,
                          float* __restrict__ gw, float* __restrict__ scol, float* __restrict__ ccol){
  int n = blockIdx.x * blockDim.x + threadIdx.x;
  if (n >= HH) return;
  float sa = 0.f, ca = 0.f;
  for (int k = 0; k < 768; k++){
    float w = W1[(size_t)k * HH + n];
    gw[(size_t)k * HH + n] = g0[k] * w;
    sa += g0[k] * w;
    ca += b0[k] * w;
  }
  scol[n] = sa; ccol[n] = ca + b1[n];
}

/* M2 = edge_emb(8x256) @ ee.lin.w(256x256) */
__global__ void k_prep_m2(const float* __restrict__ E, const float* __restrict__ Wl,
                          float* __restrict__ M2){
  int idx = blockIdx.x * blockDim.x + threadIdx.x;
  if (idx >= NEE * HH) return;
  int e = idx / HH, n = idx % HH;
  float a = 0.f;
  for (int c = 0; c < HH; c++) a += E[e*HH + c] * Wl[(size_t)c * HH + n];
  M2[idx] = a;
}

/* ---------------- heavy GEMM: h1 = LN0(ef) @ W1' , A built on the fly --------
   16(M)x64(N) tile per wave, K=768. ef = [q | k | q*k/sqrt] per position.    */
__global__ void __launch_bounds__(32) k_edge_gemm(
    const float* __restrict__ concepts, const float* __restrict__ gw,
    const float* __restrict__ scol, const float* __restrict__ ccol,
    const float* __restrict__ stats, float* __restrict__ h1){
  int lane = threadIdx.x;
  int half = lane >> 4, mr = lane & 15;
  int posb = blockIdx.x * 16;
  int tn0  = blockIdx.y * 64;
  int pos = posb + mr;
  int b = pos / (TT*TT); int r = pos % (TT*TT); int t = r / TT; int s = r % TT;
  const float* q  = concepts + (size_t)(b*TT + t) * HH;
  const float* kx = concepts + (size_t)(b*TT + s) * HH;
  v8f acc0 = {}, acc1 = {}, acc2 = {}, acc3 = {};
  for (int k0 = 0; k0 < 768; k0 += 4){
    int kk = k0 + 2*half;
    int c = kk & 255; int seg = kk >> 8;
    v2f q2 = *(const v2f*)(q + c);
    v2f k2 = *(const v2f*)(kx + c);
    v2f a;
    if (seg == 0)      a = q2;
    else if (seg == 1) a = k2;
    else               a = q2 * k2 * INV_SCALE;
    const float* wr0 = gw + (size_t)kk * HH;
    const float* wr1 = gw + (size_t)(kk + 1) * HH;
    v2f b0; b0.x = wr0[tn0      + mr]; b0.y = wr1[tn0      + mr];
    v2f b1; b1.x = wr0[tn0 + 16 + mr]; b1.y = wr1[tn0 + 16 + mr];
    v2f b2; b2.x = wr0[tn0 + 32 + mr]; b2.y = wr1[tn0 + 32 + mr];
    v2f b3; b3.x = wr0[tn0 + 48 + mr]; b3.y = wr1[tn0 + 48 + mr];
    acc0 = __builtin_amdgcn_wmma_f32_16x16x4_f32(false, a, false, b0, (short)0, acc0, false, false);
    acc1 = __builtin_amdgcn_wmma_f32_16x16x4_f32(false, a, false, b1, (short)0, acc1, false, false);
    acc2 = __builtin_amdgcn_wmma_f32_16x16x4_f32(false, a, false, b2, (short)0, acc2, false, false);
    acc3 = __builtin_amdgcn_wmma_f32_16x16x4_f32(false, a, false, b3, (short)0, acc3, false, false);
  }
  for (int i = 0; i < 8; i++){
    int rr = posb + i + 8*half;
    float mu = stats[2*rr], rs = stats[2*rr + 1];
    float mrs = mu * rs;
    int c0 = tn0 + mr;
    float* orow = h1 + (size_t)rr * HH;
    orow[c0]      = rs * acc0[i] - mrs * scol[c0]      + ccol[c0];
    orow[c0 + 16] = rs * acc1[i] - mrs * scol[c0 + 16] + ccol[c0 + 16];
    orow[c0 + 32] = rs * acc2[i] - mrs * scol[c0 + 32] + ccol[c0 + 32];
    orow[c0 + 48] = rs * acc3[i] - mrs * scol[c0 + 48] + ccol[c0 + 48];
  }
}

/* ------- fused edge tail per position (in place h1 -> edges) -------
   gelu -> ln1 -> lin2(256->8) -> softmax -> probs@M2 + bl -> ee.ln            */
__global__ void k_edge_epi(float* __restrict__ h1,
    const float* __restrict__ g1, const float* __restrict__ b1,
    const float* __restrict__ W2, const float* __restrict__ b2,
    const float* __restrict__ M2, const float* __restrict__ bl,
    const float* __restrict__ g2, const float* __restrict__ b2n){
  int pos = blockIdx.x * blockDim.y + threadIdx.y;
  if (pos >= NPOS) return;
  int lane = threadIdx.x;
  float* row = h1 + (size_t)pos * HH;
  float hv[8];
  float sm = 0.f, ss = 0.f;
  for (int j = 0; j < 8; j++){
    float x = row[lane + 32*j];
    float gch = gelu_exact(x);
    hv[j] = gch; sm += gch; ss += gch*gch;
  }
  sm = wred_sum(sm); ss = wred_sum(ss);
  float mu = sm / 256.f;
  float rstd = rsqrtf(ss / 256.f - mu*mu + 1e-5f);
  float la[8] = {0,0,0,0,0,0,0,0};
  for (int j = 0; j < 8; j++){
    int c = lane + 32*j;
    float y = (hv[j] - mu) * rstd * g1[c] + b1[c];
    const float* w = W2 + (size_t)c * NEE;
    for (int e = 0; e < 8; e++) la[e] += y * w[e];
  }
  float lg[8]; float mx = -1e30f;
  for (int e = 0; e < 8; e++){ lg[e] = wred_sum(la[e]) + b2[e]; mx = fmaxf(mx, lg[e]); }
  float se = 0.f;
  for (int e = 0; e < 8; e++){ lg[e] = expf(lg[e] - mx); se += lg[e]; }
  float inv = 1.f / se;
  float ov[8]; float sm2 = 0.f, ss2 = 0.f;
  for (int j = 0; j < 8; j++){
    int c = lane + 32*j;
    float a = bl[c];
    for (int e = 0; e < 8; e++) a += lg[e] * inv * M2[e*HH + c];
    ov[j] = a; sm2 += a; ss2 += a*a;
  }
  sm2 = wred_sum(sm2); ss2 = wred_sum(ss2);
  float mu2 = sm2 / 256.f;
  float rstd2 = rsqrtf(ss2 / 256.f - mu2*mu2 + 1e-5f);
  for (int j = 0; j < 8; j++){
    int c = lane + 32*j;
    row[c] = (ov[j] - mu2) * rstd2 * g2[c] + b2n[c];
  }
}

/* ---------------- elementwise helpers ---------------- */
__global__ void k_copy(float* o, const float* a, int n){
  int i = blockIdx.x*blockDim.x + threadIdx.x; if (i < n) o[i] = a[i];
}
__global__ void k_add(float* o, const float* a, const float* b, int n){
  int i = blockIdx.x*blockDim.x + threadIdx.x; if (i < n) o[i] = a[i] + b[i];
}
__global__ void k_add_scaled(float* o, const float* a, const float* b, float al, int n){
  int i = blockIdx.x*blockDim.x + threadIdx.x; if (i < n) o[i] = a[i] + al * b[i];
}
__global__ void k_add_sig(float* o, const float* a, const float* b, const float* sp, int n){
  int i = blockIdx.x*blockDim.x + threadIdx.x; if (i >= n) return;
  float lf = 1.f / (1.f + expf(-sp[0]));
  o[i] = a[i] + lf * b[i];
}
__global__ void k_gelu(float* o, const float* a, int n){
  int i = blockIdx.x*blockDim.x + threadIdx.x; if (i < n) o[i] = gelu_exact(a[i]);
}
__global__ void k_concat2(float* o, const float* f, const float* emb, const int* ids, int n){
  int i = blockIdx.x*blockDim.x + threadIdx.x; if (i >= n) return;
  int pos = i / 512, c = i % 512;
  o[i] = (c < 256) ? f[pos*HH + c] : emb[(size_t)ids[pos]*HH + (c - 256)];
}
__global__ void k_concat3(float* o, const float* x, const float* c1, const float* c2, int n){
  int i = blockIdx.x*blockDim.x + threadIdx.x; if (i >= n) return;
  int pos = i / 768, c = i % 768;
  o[i] = (c < 256) ? x[pos*HH + c] : (c < 512 ? c1[pos*HH + c - 256] : c2[pos*HH + c - 512]);
}

/* ======================= driver ======================= */
extern "C" void kernel_launch(void* const* d_in, const int* in_sizes, int n_in,
                              void* d_out, int out_size, void* d_ws, size_t ws_size,
                              hipStream_t stream){
  (void)in_sizes; (void)n_in; (void)out_size; (void)ws_size;
  const float* x  = (const float*)d_in[0];
  const int* ids  = (const int*)d_in[1];
#define PF(i) ((const float*)d_in[i])
  /* param leaf order (alphabetical pytree):
     2:ch_norm.b 3:ch_norm.g 4:cproj.b 5:cproj.w
     6..13: cross[0..1]{bo,bqkv,wo,wqkv}  14:edge_emb
     15:ee.lin.b 16:ee.lin.w 17:ee.ln.b 18:ee.ln.g
     19:es.lin1.b 20:es.lin1.w 21:es.lin2.b 22:es.lin2.w 23:es.ln0.b 24:es.ln0.g 25:es.ln1.b 26:es.ln1.g
     27..38: enhancers[0..2]{lin.b,lin.w,ln.b,ln.g}
     39..74: extractors[0..2]{attn.bo,attn.bqkv,attn.wo,attn.wqkv,ff1.b,ff1.w,ff2.b,ff2.w,ln1.b,ln1.g,ln2.b,ln2.g}
     75:gb_factor 76..87: graph_layers[0..1]{bo,bqkv,wo,wqkv,ln.b,ln.g}
     88:gproj.b 89:gproj.w 90..92:level_concepts 93..95:level_factors
     96:sem.b 97:sem.w 98:sem_ln.b 99:sem_ln.g */

  float* WS = (float*)d_ws;
  size_t off = 0;
  auto alloc = [&](size_t nf){ float* p = WS + off; off += (nf + 63) & ~(size_t)63; return p; };
  const int NT = ROWS * HH;
  float* cur     = alloc(NT);
  float* tmp1    = alloc(NT);
  float* qkvb    = alloc(ROWS * 768);
  float* attn_o  = alloc(NT);
  float* mha_o   = alloc(NT);
  float* hbuf    = alloc(NT);
  float* ffmid   = alloc(ROWS * 1024);
  float* feats   = alloc(NT);
  float* cat2    = alloc(ROWS * 512);
  float* enh     = alloc(NT);
  float* acc     = alloc(NT);
  float* cpts    = alloc(NT);
  float* gcur    = alloc(NT);
  float* graphb  = alloc(NT);
  float* cat3    = alloc(ROWS * 768);
  float* semb    = alloc(NT);
  float* gw1     = alloc(768 * HH);
  float* scol    = alloc(HH);
  float* ccol    = alloc(HH);
  float* m2      = alloc(NEE * HH);
  float* stats   = alloc(2 * (size_t)NPOS);
  float* edges   = alloc((size_t)NPOS * HH);   /* ~75.5 MB */

  dim3 eb(256);
  auto eg = [](int n){ return dim3((unsigned)((n + 255) / 256)); };
  dim3 lnB(32, 8), lnG(48);

  k_copy<<<eg(NT), eb, 0, stream>>>(cur, x, NT);

  for (int lvl = 0; lvl < 3; lvl++){
    int ex = 39 + lvl*12, en = 27 + lvl*4;
    /* encoder: h = cur + mha(ln1(cur)) */
    k_layernorm<<<lnG, lnB, 0, stream>>>(cur, PF(ex+9), PF(ex+8), tmp1, ROWS, HH);
    k_gemm16<<<dim3(48,24), 32, 0, stream>>>(tmp1, PF(ex+3), PF(ex+1), qkvb, ROWS, 768, 256, 0);
    k_attn<<<dim3(TT,NHH,BB), 32, 0, stream>>>(qkvb, attn_o, nullptr, nullptr);
    k_gemm16<<<dim3(16,24), 32, 0, stream>>>(attn_o, PF(ex+2), PF(ex+0), mha_o, ROWS, 256, 256, 0);
    k_add<<<eg(NT), eb, 0, stream>>>(hbuf, cur, mha_o, NT);
    /* feats = h + ff2(relu(ff1(ln2(h)))) */
    k_layernorm<<<lnG, lnB, 0, stream>>>(hbuf, PF(ex+11), PF(ex+10), tmp1, ROWS, HH);
    k_gemm16<<<dim3(64,24), 32, 0, stream>>>(tmp1, PF(ex+5), PF(ex+4), ffmid, ROWS, 1024, 256, 1);
    k_gemm16<<<dim3(16,24), 32, 0, stream>>>(ffmid, PF(ex+7), PF(ex+6), mha_o, ROWS, 256, 1024, 0);
    k_add<<<eg(NT), eb, 0, stream>>>(feats, hbuf, mha_o, NT);
    /* enhancer */
    k_concat2<<<eg(ROWS*512), eb, 0, stream>>>(cat2, feats, PF(90+lvl), ids, ROWS*512);
    k_gemm16<<<dim3(16,24), 32, 0, stream>>>(cat2, PF(en+1), PF(en+0), tmp1, ROWS, 256, 512, 0);
    k_layernorm<<<lnG, lnB, 0, stream>>>(tmp1, PF(en+3), PF(en+2), enh, ROWS, HH);
    k_gelu<<<eg(NT), eb, 0, stream>>>(enh, enh, NT);
    k_add_sig<<<eg(NT), eb, 0, stream>>>(tmp1, cur, enh, PF(93+lvl), NT);
    k_layernorm<<<lnG, lnB, 0, stream>>>(tmp1, PF(3), PF(2), cur, ROWS, HH);
    if (lvl == 0) k_copy<<<eg(NT), eb, 0, stream>>>(acc, cur, NT);
    else          k_add<<<eg(NT), eb, 0, stream>>>(acc, acc, cur, NT);
    if (lvl < 2){
      int cb = 6 + lvl*4;
      k_gemm16<<<dim3(48,24), 32, 0, stream>>>(cur, PF(cb+3), PF(cb+1), qkvb, ROWS, 768, 256, 0);
      k_attn<<<dim3(TT,NHH,BB), 32, 0, stream>>>(qkvb, attn_o, nullptr, nullptr);
      k_gemm16<<<dim3(16,24), 32, 0, stream>>>(attn_o, PF(cb+2), PF(cb+0), mha_o, ROWS, 256, 256, 0);
      k_add_scaled<<<eg(NT), eb, 0, stream>>>(cur, cur, mha_o, 0.1f, NT);
    }
  }

  /* concepts = (x + 0.1*mean(allc)) @ concept_proj */
  k_add_scaled<<<eg(NT), eb, 0, stream>>>(tmp1, x, acc, 0.1f/3.0f, NT);
  k_gemm16<<<dim3(16,24), 32, 0, stream>>>(tmp1, PF(5), PF(4), cpts, ROWS, 256, 256, 0);

  /* edges */
  k_prep_w1<<<dim3(1), dim3(256), 0, stream>>>(PF(20), PF(24), PF(23), PF(19), gw1, scol, ccol);
  k_prep_m2<<<dim3(8), dim3(256), 0, stream>>>(PF(14), PF(16), m2);
  k_edge_stats<<<dim3(NPOS/8), dim3(32,8), 0, stream>>>(cpts, stats);
  k_edge_gemm<<<dim3(NPOS/16, 4), 32, 0, stream>>>(cpts, gw1, scol, ccol, stats, edges);
  k_edge_epi<<<dim3(NPOS/4), dim3(32,4), 0, stream>>>(edges, PF(26), PF(25), PF(22), PF(21),
                                                      m2, PF(15), PF(18), PF(17));

  /* graph layers with edge bias */
  k_copy<<<eg(NT), eb, 0, stream>>>(gcur, cpts, NT);
  for (int gl = 0; gl < 2; gl++){
    int gb = 76 + gl*6;
    k_gemm16<<<dim3(48,24), 32, 0, stream>>>(gcur, PF(gb+3), PF(gb+1), qkvb, ROWS, 768, 256, 0);
    k_attn<<<dim3(TT,NHH,BB), 32, 0, stream>>>(qkvb, attn_o, gcur, edges);
    k_gemm16<<<dim3(16,24), 32, 0, stream>>>(attn_o, PF(gb+2), PF(gb+0), mha_o, ROWS, 256, 256, 0);
    k_add<<<eg(NT), eb, 0, stream>>>(tmp1, gcur, mha_o, NT);
    k_layernorm<<<lnG, lnB, 0, stream>>>(tmp1, PF(gb+5), PF(gb+4), gcur, ROWS, HH);
  }

  /* graph proj, semantic head */
  k_add_sig<<<eg(NT), eb, 0, stream>>>(tmp1, cpts, gcur, PF(75), NT);
  k_gemm16<<<dim3(16,24), 32, 0, stream>>>(tmp1, PF(89), PF(88), graphb, ROWS, 256, 256, 0);
  k_concat3<<<eg(ROWS*768), eb, 0, stream>>>(cat3, x, cpts, graphb, ROWS*768);
  k_gemm16<<<dim3(16,24), 32, 0, stream>>>(cat3, PF(97), PF(96), tmp1, ROWS, 256, 768, 0);
  k_layernorm<<<lnG, lnB, 0, stream>>>(tmp1, PF(99), PF(98), semb, ROWS, HH);
  k_gelu<<<eg(NT), eb, 0, stream>>>(semb, semb, NT);
  k_add_scaled<<<eg(NT), eb, 0, stream>>>((float*)d_out, x, semb, 0.1f, NT);
#undef PF
}